// Filter_param_22926535426501
// MI455X (gfx1250) — compile-verified
//
#include <hip/hip_runtime.h>
#include <math.h>

// Problem constants (fixed by the reference).
#define T_LEN 8192
#define B_ROWS 2048
// GEMV tiling.
#define KSEG 8                       // K split across gridDim.y
#define SEG_LEN (T_LEN / KSEG)       // 1024 columns per segment
#define ROWS_PER_WG 128              // 8 waves * 16-row WMMA tiles
#define KC 64                        // columns staged in LDS per chunk
#define LDS_STRIDE (KC + 4)          // 68 floats: 272B row stride -> 16B aligned, bank shift 4
#define NCHUNK (SEG_LEN / KC)        // 16 chunks per segment

typedef float v2f __attribute__((ext_vector_type(2)));
typedef float v8f __attribute__((ext_vector_type(8)));

struct FiltParams { float kexp, fs, fi, cc, wo, iwT; };

__device__ __forceinline__ float sigmoidf_(float v) { return 1.0f / (1.0f + __expf(-v)); }
__device__ __forceinline__ float squash_(float v, float lo, float hi) { return lo + (hi - lo) * sigmoidf_(v); }

__device__ __forceinline__ void abw_(const FiltParams& P, int t, float& a, float& b, float& w) {
  float tn = (float)t * (1.0f / (float)T_LEN);
  float f = P.fi + (P.fs - P.fi) * __expf(P.kexp * tn);
  a = (f - P.cc) / (f + P.cc);
  b = 0.5f * (a + 1.0f);
  w = sigmoidf_((tn - P.wo) * P.iwT);
}

// ---------------------------------------------------------------------------
// Phase 1: single workgroup. Chunked-parallel backward affine scan
//   g_u = g_{u+1} * A_{u+1} + w_u * e2   (row-vector form, 2x2 A)
// then emit d[t] (GEMV coefficient vector) and 1/sum(w).
// ---------------------------------------------------------------------------
__global__ __launch_bounds__(256) void filt_coef_kernel(
    const int* __restrict__ Fp,
    const float* __restrict__ fsn, const float* __restrict__ fin,
    const float* __restrict__ fdn, const float* __restrict__ ftn,
    const float* __restrict__ won, const float* __restrict__ wtn,
    float* __restrict__ coef, float* __restrict__ invW)
{
  __shared__ float sM00[256], sM01[256], sM10[256], sM11[256], sp1[256], sp2[256];
  __shared__ float sg1[256], sg2[256];
  __shared__ float wred[256];
  __shared__ float cbuf[T_LEN];          // 32KB: c_u coefficients
  __shared__ float shQ;

  const int tid = threadIdx.x;
  FiltParams P;
  {
    float F = (float)(*Fp);
    P.cc = sqrtf(exp2f(1.0f / 3.0f) - 1.0f) * F / 3.14159265358979323846f;
    P.fs = squash_(*fsn, 200.0f, 4000.0f);
    P.fi = squash_(*fin, 20.0f, 500.0f);
    float fd = squash_(*fdn, 0.0f, 2.0f);
    float fT = squash_(*ftn, 0.0f, 2.0f);
    P.wo = squash_(*won, 0.0f, 1.0f);
    float wT = squash_(*wtn, 0.01f, 0.5f);
    float ln10 = logf(10.0f);
    P.kexp = (logf(0.7f) - fd * ln10) * (0.1f * __expf(fT * ln10));
    P.iwT = 1.0f / wT;
  }
  const int CH = T_LEN / 256;            // 32 steps per lane
  const int lo = tid * CH, hi = lo + CH;

  // Pass 1: per-chunk affine map (M, p) mapping g_hi -> g_lo; also partial sum(w).
  float M00 = 1.f, M01 = 0.f, M10 = 0.f, M11 = 1.f, p1 = 0.f, p2 = 0.f, wsum = 0.f;
  for (int u = hi - 1; u >= lo; --u) {
    float au, bu, wu; abw_(P, u, au, bu, wu);
    wsum += wu;
    if (u == 0) break;                   // scan domain is u >= 1
    float an, bn, wn; abw_(P, u + 1, an, bn, wn);
    float na = -an, bb = bn * (1.0f - an);
    float n00 = M00 * na + M01 * bb, n01 = M01 * na;
    float n10 = M10 * na + M11 * bb, n11 = M11 * na;
    float q1 = p1 * na + p2 * bb,    q2 = p2 * na + wu;
    M00 = n00; M01 = n01; M10 = n10; M11 = n11; p1 = q1; p2 = q2;
  }
  sM00[tid] = M00; sM01[tid] = M01; sM10[tid] = M10; sM11[tid] = M11;
  sp1[tid] = p1; sp2[tid] = p2; wred[tid] = wsum;
  __syncthreads();

  // Serial combine over 256 chunk maps (top to bottom) -> seed g at each chunk's hi.
  if (tid == 0) {
    float g1 = 0.f, g2 = 0.f;
    for (int i = 255; i >= 0; --i) {
      sg1[i] = g1; sg2[i] = g2;
      float n1 = g1 * sM00[i] + g2 * sM10[i] + sp1[i];
      float n2 = g1 * sM01[i] + g2 * sM11[i] + sp2[i];
      g1 = n1; g2 = n2;
    }
  }
  __syncthreads();

  // Replay: emit c_u = g_u1*b_u + g_u2*b_u^2.
  {
    float g1 = sg1[tid], g2 = sg2[tid];
    const int stop = (lo < 1) ? 1 : lo;
    for (int u = hi - 1; u >= stop; --u) {
      float an, bn, wn; abw_(P, u + 1, an, bn, wn);
      float au, bu, wu; abw_(P, u, au, bu, wu);
      float na = -an, bb = bn * (1.0f - an);
      float n1 = g1 * na + g2 * bb;
      float n2 = g2 * na + wu;
      g1 = n1; g2 = n2;
      cbuf[u] = g1 * bu + g2 * bu * bu;
    }
    if (tid == 0) {                       // q^T = g_1^T A_1 ; s0 = (x0,x0); plus w0 term
      float a1, b1, w1; abw_(P, 1, a1, b1, w1);
      float a0, b0, w0; abw_(P, 0, a0, b0, w0);
      shQ = w0 + g1 * (-a1) + g2 * (b1 * (1.0f - a1)) + g2 * (-a1);
    }
  }
  __syncthreads();

  // sum(w) reduction.
  for (int s = 128; s > 0; s >>= 1) {
    if (tid < s) wred[tid] += wred[tid + s];
    __syncthreads();
  }
  if (tid == 0) invW[0] = 1.0f / wred[0];

  // Final coefficients: out[b] = sum_t d[t]*X[b,t] * invW.
  for (int t = lo; t < hi; ++t) {
    float dt;
    if (t == 0)              dt = shQ + cbuf[1];
    else if (t == T_LEN - 1) dt = cbuf[T_LEN - 1];
    else                     dt = cbuf[t] + cbuf[t + 1];
    coef[t] = dt;
  }
}

// ---------------------------------------------------------------------------
// Phase 2: GEMV via V_WMMA_F32_16X16X4_F32, fed by double-buffered
// GLOBAL_LOAD_ASYNC_TO_LDS_B128 (ASYNCcnt) so HBM streaming overlaps WMMA.
// Each wave owns a 16-row tile; B operand broadcasts coef; column 0 extracted.
// ---------------------------------------------------------------------------
__global__ __launch_bounds__(256) void filt_gemv_wmma_kernel(
    const float* __restrict__ X, const float* __restrict__ coef,
    float* __restrict__ part)
{
  __shared__ float tile[2][ROWS_PER_WG * LDS_STRIDE]; // 2 x 34 KB
  __shared__ float cf[2][KC];                         // 2 x 256 B

  const int tid = threadIdx.x;
  const int wave = tid >> 5, lane = tid & 31;
  const int rowbase = blockIdx.x * ROWS_PER_WG;
  const int segbase = blockIdx.y * SEG_LEN;
  const int rl   = wave * 16 + (lane & 15);  // local row for A operand
  const int koff = (lane >> 4) * 2;          // K pair select (lanes 16-31 -> K=2,3)

  // Per-thread async-load mapping: 16 threads * 16B per row, 16 rows per pass.
  const int lr = tid >> 4;                   // row within pass group (0..15)
  const int lc = (tid & 15) * 4;             // column (floats)
  const unsigned voff_base = (unsigned)(lr * T_LEN + lc) * 4u;       // global byte offset
  const unsigned lds_off_base = (unsigned)(lr * LDS_STRIDE + lc) * 4u;
  // LDS byte addresses = low 32 bits of the generic pointer (aperture is in [63:32]).
  const unsigned tile_lds0 = (unsigned)(uintptr_t)&tile[0][0];
  const unsigned tile_lds1 = (unsigned)(uintptr_t)&tile[1][0];
  const unsigned cf_lds0   = (unsigned)(uintptr_t)&cf[0][0];
  const unsigned cf_lds1   = (unsigned)(uintptr_t)&cf[1][0];
  const float* xbase = X + (size_t)rowbase * T_LEN;

  auto issue_chunk = [&](int buf, int kb) {
    const unsigned tbase = (buf ? tile_lds1 : tile_lds0);
    // cf first (wave 0 only): keeps wave0's oldest-9 = one full chunk.
    if (tid < 16) {
      unsigned dst = (buf ? cf_lds1 : cf_lds0) + (unsigned)tid * 16u;
      unsigned vo  = (unsigned)tid * 16u;
      asm volatile("global_load_async_to_lds_b128 %0, %1, %2"
                   :: "v"(dst), "v"(vo), "s"(coef + kb) : "memory");
    }
    const float* sbase = xbase + kb;
#pragma unroll
    for (int pass = 0; pass < ROWS_PER_WG / 16; ++pass) {
      unsigned dst = tbase + lds_off_base + (unsigned)(pass * 16 * LDS_STRIDE * 4);
      unsigned vo  = voff_base + (unsigned)(pass * 16 * T_LEN * 4);
      asm volatile("global_load_async_to_lds_b128 %0, %1, %2"
                   :: "v"(dst), "v"(vo), "s"(sbase) : "memory");
    }
  };

  v8f acc = {};
  issue_chunk(0, segbase);                       // prologue: chunk 0 -> buf 0
  for (int chunk = 0; chunk < NCHUNK; ++chunk) {
    const int buf = chunk & 1;
    if (chunk + 1 < NCHUNK) {
      issue_chunk(buf ^ 1, segbase + (chunk + 1) * KC);
      // Async loads retire in order: <=8 outstanding => all of chunk's 8 (9 on
      // wave0, which issued 9+9=18 and retires down to 8 -> 10 done) landed.
      asm volatile("s_wait_asynccnt 0x8" ::: "memory");
    } else {
      asm volatile("s_wait_asynccnt 0x0" ::: "memory");
    }
    __syncthreads();                             // all waves' LDS writes visible
    const float* tl = &tile[buf][0];
    const float* cl = &cf[buf][0];
#pragma unroll
    for (int k = 0; k < KC; k += 4) {
      v2f av = *reinterpret_cast<const v2f*>(&tl[rl * LDS_STRIDE + k + koff]);
      v2f bv = *reinterpret_cast<const v2f*>(&cl[k + koff]);
      // D = A(16x4 f32) * B(4x16 f32) + C
      acc = __builtin_amdgcn_wmma_f32_16x16x4_f32(
          false, av, false, bv, (short)0, acc, false, false);
    }
    __syncthreads();                             // buffer reusable for chunk+2
  }

  // C layout: VGPR r, lane L -> M = r + 8*(L>>4), N = L&15. Take N==0 column.
  if ((lane & 15) == 0) {
    const int rbase = rowbase + wave * 16 + ((lane >> 4) << 3);
    float* dst = part + (size_t)blockIdx.y * B_ROWS + rbase;
#pragma unroll
    for (int r = 0; r < 8; ++r) dst[r] = acc[r];
  }
}

// ---------------------------------------------------------------------------
// Phase 3: deterministic reduction over K segments + 1/sum(w) scaling.
// ---------------------------------------------------------------------------
__global__ __launch_bounds__(256) void filt_reduce_kernel(
    const float* __restrict__ part, const float* __restrict__ invW,
    float* __restrict__ out)
{
  const int i = blockIdx.x * 256 + threadIdx.x;
  if (i < B_ROWS) {
    float s = 0.f;
#pragma unroll
    for (int k = 0; k < KSEG; ++k) s += part[(size_t)k * B_ROWS + i];
    out[i] = s * invW[0];
  }
}

extern "C" void kernel_launch(void* const* d_in, const int* in_sizes, int n_in,
                              void* d_out, int out_size, void* d_ws, size_t ws_size,
                              hipStream_t stream) {
  const float* X   = (const float*)d_in[0];
  const int*   Fp  = (const int*)  d_in[1];
  const float* fsn = (const float*)d_in[2];
  const float* fin = (const float*)d_in[3];
  const float* fdn = (const float*)d_in[4];
  const float* ftn = (const float*)d_in[5];
  const float* won = (const float*)d_in[6];
  const float* wtn = (const float*)d_in[7];

  // Workspace layout (floats): [0,8192) coef d[t]; [8192] invW; [8256, +KSEG*B) partials.
  float* coef = (float*)d_ws;
  float* invW = coef + T_LEN;
  float* part = coef + T_LEN + 64;

  filt_coef_kernel<<<1, 256, 0, stream>>>(Fp, fsn, fin, fdn, ftn, won, wtn, coef, invW);
  filt_gemv_wmma_kernel<<<dim3(B_ROWS / ROWS_PER_WG, KSEG), 256, 0, stream>>>(X, coef, part);
  filt_reduce_kernel<<<(B_ROWS + 255) / 256, 256, 0, stream>>>(part, invW, (float*)d_out);
}